// LengthRegulator_21912923144299
// MI455X (gfx1250) — compile-verified
//
#include <hip/hip_runtime.h>
#include <cstdint>

// Problem constants from the reference: B=32, T=1024, D=1024, N_ITER=256, SR=4
#define B_SZ   32
#define T_SZ   1024
#define D_SZ   1024
#define N_IT   256
#define CHUNK  8          // ODE steps staged per LDS tile
#define PITCH  9          // 8 d-values + 1 pad dword per t-row (TDM pad_enable) -> bank-conflict-free gather
#define NCHUNK (N_IT / CHUNK)

typedef unsigned int v4u __attribute__((ext_vector_type(4)));
typedef int          v8i __attribute__((ext_vector_type(8)));
typedef int          v4i __attribute__((ext_vector_type(4)));
typedef float        v4f __attribute__((ext_vector_type(4)));

#if __has_builtin(__builtin_amdgcn_tensor_load_to_lds) && __has_builtin(__builtin_amdgcn_s_wait_tensorcnt)
#define HAVE_TDM 1
#else
#define HAVE_TDM 0
#endif

#if HAVE_TDM
// Issue one TDM DMA: ds[b, 0:1024, n0:n0+8] -> LDS tile (pitch 9 dwords via pad_enable).
__device__ __forceinline__ void tdm_issue_chunk(const float* gsrc, float* lds_tile) {
  uint64_t ga     = (uint64_t)(uintptr_t)gsrc;
  uint32_t lds_lo = (uint32_t)(uintptr_t)lds_tile;     // generic LDS addr low 32b = LDS offset
  // ---- D# group 0: count=1, lds_addr, global_addr[56:0], type=2 ----
  v4u g0;
  g0.x = 1u;
  g0.y = lds_lo;
  g0.z = (uint32_t)ga;
  g0.w = (uint32_t)((ga >> 32) & 0x1FFFFFFu) | (2u << 30);
  // ---- D# group 1: data_size=4B, pad 1 dword every 8 dwords,
  //      tensor_dim0=256 (n, contiguous), tensor_dim1=1024 (t),
  //      tile = 8 x 1024, tensor_dim0_stride = 256 ----
  v8i g1;
  g1[0] = (int)((2u << 16) | (1u << 20) | (2u << 22));
  g1[1] = (int)((uint32_t)N_IT << 16);   // tensor_dim0[15:0]
  g1[2] = (int)((uint32_t)T_SZ << 16);   // tensor_dim1[15:0]
  g1[3] = (int)((uint32_t)CHUNK << 16);  // tile_dim0 = 8
  g1[4] = (int)((uint32_t)T_SZ);         // tile_dim1 = 1024
  g1[5] = (int)((uint32_t)N_IT);         // tensor_dim0_stride = 256
  g1[6] = 0;
  g1[7] = 0;
  v4i z4 = {0, 0, 0, 0};                 // groups 2/3 unused (2D tensor)
  v8i z8 = {0, 0, 0, 0, 0, 0, 0, 0};
  __builtin_amdgcn_tensor_load_to_lds(g0, g1, z4, z4, z8, 0);
}
#endif

// ---------------------------------------------------------------------------
// Phase 1: sequential warp ODE.  grid = B*4 blocks (each block owns 256 t's of
// one batch), 256 threads (8 wave32), one t per thread held in a register.
// Chunks of ds[b, :, n0:n0+8] are DMA'd into LDS by the Tensor Data Mover,
// double-buffered: chunk k+1 streams in (TENSORcnt<=1) while the 8 Euler steps
// of chunk k run out of LDS.  TENSORcnt completes in-order per wave, so
// waiting <=1 after issuing k+1 guarantees chunk k is resident.
// ---------------------------------------------------------------------------
__global__ __launch_bounds__(256) void gw_ode_kernel(const float* __restrict__ ds,
                                                     float* __restrict__ f_out) {
  __shared__ float tile[2][T_SZ * PITCH];   // 2 x 36 KB (WGP has 320 KB)
  const int b = blockIdx.x >> 2;
  const int q = blockIdx.x & 3;
  const int t = (q << 8) + threadIdx.x;
  const float* bsrc = ds + (uint64_t)b * T_SZ * N_IT;

  float f = (float)t;                       // identity grid init (step 0 == reference f0)

#if HAVE_TDM
  if ((threadIdx.x >> 5) == 0) {            // wave 0 owns the DMA pipeline
    tdm_issue_chunk(bsrc, tile[0]);         // prologue: chunk 0
  }
#endif

  for (int k = 0; k < NCHUNK; ++k) {
    float* cur = tile[k & 1];
#if HAVE_TDM
    if ((threadIdx.x >> 5) == 0) {
      if (k + 1 < NCHUNK) {
        tdm_issue_chunk(bsrc + (uint64_t)(k + 1) * CHUNK, tile[(k + 1) & 1]);
        __builtin_amdgcn_s_wait_tensorcnt(1);   // chunk k resident; k+1 in flight
      } else {
        __builtin_amdgcn_s_wait_tensorcnt(0);
      }
    }
#else
    // Fallback: cooperative staged copy (32 B contiguous runs per row)
    for (int i = threadIdx.x; i < T_SZ * CHUNK; i += 256) {
      int tt = i >> 3, dn = i & 7;
      cur[tt * PITCH + dn] = bsrc[(uint64_t)tt * N_IT + (uint64_t)(k * CHUNK + dn)];
    }
#endif
    __syncthreads();

#pragma unroll
    for (int dn = 0; dn < CHUNK; ++dn) {
      int i0 = (int)f;                      // f >= 0 always -> trunc == floor
      if (i0 > T_SZ - 2) i0 = T_SZ - 2;
      float fr = f - (float)i0;
      float d0 = cur[i0 * PITCH + dn];      // lanes hit distinct banks (pitch 9 coprime 64)
      float d1 = cur[i0 * PITCH + PITCH + dn];
      f += (d0 + fr * (d1 - d0)) * (1.0f / (float)(N_IT * 4));   // ds / (N*SR)
      f = fminf(fmaxf(f, 0.0f), (float)(T_SZ - 1));
    }
    __syncthreads();                        // all reads of `cur` done before it is re-filled
  }

  f_out[(uint64_t)b * T_SZ + t] = f;
}

// ---------------------------------------------------------------------------
// Phase 2: Catmull-Rom cubic resample.  One block per (b, t); the 4 source
// rows are wave-uniform, lanes stride channels with 128-bit (v4f) accesses.
// HBM-bound: ~134 MB in + 134 MB out.  Output stores are non-temporal so the
// write-once ys stream does not evict xs rows (re-read ~4x by neighboring t
// blocks) from the 192 MB L2.
// ---------------------------------------------------------------------------
__global__ __launch_bounds__(256) void cubic_kernel(const float* __restrict__ xs,
                                                    const float* __restrict__ fw,
                                                    float* __restrict__ ys) {
  const int bt = blockIdx.x;                // b*T + t
  const int b  = bt >> 10;

  const float fv = fw[bt];                  // wave-uniform broadcast load
  int i1 = (int)fv;
  if (i1 < 0) i1 = 0;
  if (i1 > T_SZ - 1) i1 = T_SZ - 1;
  const float t  = fv - (float)i1;
  const float t2 = t * t;
  const float t3 = t2 * t;
  const float w0 = -0.5f * t3 + t2 - 0.5f * t;
  const float w1 =  1.5f * t3 - 2.5f * t2 + 1.0f;
  const float w2 = -1.5f * t3 + 2.0f * t2 + 0.5f * t;
  const float w3 =  0.5f * t3 - 0.5f * t2;

  const int im1 = (i1 - 1 < 0)        ? 0        : i1 - 1;
  const int ip1 = (i1 + 1 > T_SZ - 1) ? T_SZ - 1 : i1 + 1;
  const int ip2 = (i1 + 2 > T_SZ - 1) ? T_SZ - 1 : i1 + 2;

  const uint64_t base = (uint64_t)b * T_SZ * D_SZ;
  const v4f* __restrict__ r0 = (const v4f*)(xs + base + (uint64_t)im1 * D_SZ);
  const v4f* __restrict__ r1 = (const v4f*)(xs + base + (uint64_t)i1  * D_SZ);
  const v4f* __restrict__ r2 = (const v4f*)(xs + base + (uint64_t)ip1 * D_SZ);
  const v4f* __restrict__ r3 = (const v4f*)(xs + base + (uint64_t)ip2 * D_SZ);
  v4f* __restrict__ o = (v4f*)(ys + (uint64_t)bt * D_SZ);

  const int c = threadIdx.x;                // 256 threads * v4f == D=1024 channels
  const v4f p0 = r0[c], p1 = r1[c], p2 = r2[c], p3 = r3[c];
  v4f r = w0 * p0 + w1 * p1 + w2 * p2 + w3 * p3;   // elementwise on ext-vector
  __builtin_nontemporal_store(r, &o[c]);    // TH=NT: keep ys out of L2
}

// ---------------------------------------------------------------------------
// Host entry. Inputs: d_in[0]=xs (B*T*D f32), d_in[1]=ds (B*T*N f32),
// d_in[2]=is_inference (unused). Output tuple flattened: ys (B*T*D) then f (B*T).
// ---------------------------------------------------------------------------
extern "C" void kernel_launch(void* const* d_in, const int* in_sizes, int n_in,
                              void* d_out, int out_size, void* d_ws, size_t ws_size,
                              hipStream_t stream) {
  (void)in_sizes; (void)n_in; (void)out_size; (void)d_ws; (void)ws_size;
  const float* xs = (const float*)d_in[0];
  const float* ds = (const float*)d_in[1];
  float* ys = (float*)d_out;
  float* f  = ys + (size_t)B_SZ * T_SZ * D_SZ;   // second tuple element

  gw_ode_kernel<<<B_SZ * 4, 256, 0, stream>>>(ds, f);
  cubic_kernel<<<B_SZ * T_SZ, 256, 0, stream>>>(xs, f, ys);
}